// KGATConv_59734405152887
// MI455X (gfx1250) — compile-verified
//
#include <hip/hip_runtime.h>
#include <hip/hip_bf16.h>

// KGATConv for MI455X (gfx1250, wave32).
// - nfeat (12.8MB) + relation_W (256KB) are L2-resident: bin edges by relation,
//   compute W_r·h_src / W_r·h_dst per 16-edge batch with fp32 WMMA (16x16x4),
//   fusing the att = sum T*tanh(H+efeat) epilogue (no 205MB proj material.).
// - One BLOCK per (relation, slice): W_r staged into LDS ONCE per block via
//   GLOBAL_LOAD_ASYNC_TO_LDS_B128 (ASYNCcnt path), shared by 4 waves looping
//   over that relation's 16-edge groups.
// - Softmax-by-dst via monotone-uint atomicMax + atomicAdd; scatter via
//   global_atomic_add_f32; final bi-interaction GEMMs via WMMA, W1/W2 in LDS.

#define DFEAT 64
#define MAXREL 16
#define BPR 64  // blocks per relation in edge_att

typedef __attribute__((ext_vector_type(2))) float v2f;
typedef __attribute__((ext_vector_type(8))) float v8f;

#if defined(__HIP_DEVICE_COMPILE__) && \
    !__has_builtin(__builtin_amdgcn_wmma_f32_16x16x4_f32)
#error "device: missing __builtin_amdgcn_wmma_f32_16x16x4_f32"
#endif

__device__ __forceinline__ v8f wmma4(v2f a, v2f b, v8f c) {
  // D = A(16x4 f32) x B(4x16 f32) + C(16x16 f32)
  return __builtin_amdgcn_wmma_f32_16x16x4_f32(false, a, false, b, (short)0, c,
                                               false, false);
}

// Async global->LDS copy, 16B per lane. Tracked by ASYNCcnt.
__device__ __forceinline__ void async_load_b128(const void* gptr,
                                                unsigned lds_off) {
  asm volatile("global_load_async_to_lds_b128 %0, %1, off"
               :
               : "v"(lds_off), "v"(gptr)
               : "memory");
}
__device__ __forceinline__ void wait_asynccnt0() {
  asm volatile("s_wait_asynccnt 0x0" ::: "memory");
}
// Generic (flat) shared-pointer -> raw LDS byte offset (ISA: low 32 bits).
__device__ __forceinline__ unsigned lds_offset(const void* p) {
  return (unsigned)(unsigned long long)p;
}

// order-preserving float -> uint mapping (for atomicMax-based segment max)
__device__ __forceinline__ unsigned fkey(float x) {
  unsigned u = __float_as_uint(x);
  return (u & 0x80000000u) ? ~u : (u | 0x80000000u);
}
__device__ __forceinline__ float funkey(unsigned k) {
  unsigned u = (k & 0x80000000u) ? (k & 0x7FFFFFFFu) : ~k;
  return __uint_as_float(u);
}

// ---------------------------------------------------------------- init
__global__ void init_kernel(float* hn, float* denom, unsigned* segkey,
                            int* cnt, int N, int R, long total) {
  long i = (long)blockIdx.x * blockDim.x + threadIdx.x;
  if (i < total) hn[i] = 0.0f;            // h_neighbor output region
  if (i < N) { denom[i] = 0.0f; segkey[i] = 0u; }
  if (i < R) cnt[i] = 0;
}

// ---------------------------------------------------------------- binning
__global__ void hist_kernel(const int* ety, int* cnt, int E) {
  int e = blockIdx.x * blockDim.x + threadIdx.x;
  if (e < E) atomicAdd(&cnt[ety[e]], 1);
}

__global__ void prefix_kernel(const int* cnt, int* bin_start, int* cursor,
                              int R) {
  if (blockIdx.x == 0 && threadIdx.x == 0) {
    int s = 0;
    for (int r = 0; r < R; ++r) {
      bin_start[r] = s;
      cursor[r] = s;
      s += cnt[r];
    }
    bin_start[R] = s;
  }
}

__global__ void scatter_kernel(const int* ety, int* cursor, int* bin_edges,
                               int E) {
  int e = blockIdx.x * blockDim.x + threadIdx.x;
  if (e < E) {
    int pos = atomicAdd(&cursor[ety[e]], 1);
    bin_edges[pos] = e;
  }
}

// ---------------------------------------------------------------- edge attention
// Block = (relation r, slice s). W_r staged once per block (async -> LDS),
// then each of the 4 waves loops over 16-edge groups of bin r with stride.
//   T = nfeat[src_rows] @ W_r   (16x64)   via WMMA f32 16x16x4
//   H = nfeat[dst_rows] @ W_r   (16x64)
//   att[e] = sum_k T[e,k] * tanh(H[e,k] + efeat[e,k])
__global__ __launch_bounds__(128) void edge_att_kernel(
    const float* __restrict__ nfeat, const float* __restrict__ efeat,
    const float* __restrict__ relW, const int* __restrict__ src,
    const int* __restrict__ dst, const int* __restrict__ bin_edges,
    const int* __restrict__ bin_start, const int* __restrict__ cnt,
    float* __restrict__ att, unsigned* __restrict__ segkey) {
  __shared__ float Ws[DFEAT * DFEAT];  // 16KB: this block's W_r

  const int r = blockIdx.x / BPR;       // relation handled by this block
  const int s = blockIdx.x - r * BPR;   // slice within relation
  const int wave = threadIdx.x >> 5;
  const int lane = threadIdx.x & 31;

  // ---- async-stage W_r into LDS (16KB = 1024 x b128; 8 per thread) ----
  {
    const float* W = relW + (size_t)r * DFEAT * DFEAT;
    const unsigned base = lds_offset(&Ws[0]);
    for (int i = threadIdx.x; i < (DFEAT * DFEAT) / 4; i += 128)
      async_load_b128(W + i * 4, base + i * 16);
    wait_asynccnt0();
  }
  __syncthreads();

  const int bstart = bin_start[r];
  const int bcnt = cnt[r];
  const int ngroups = (bcnt + 15) >> 4;

  const int h = lane >> 4;   // half-wave id
  const int c = lane & 15;   // column-within-half / A-row id
  const v2f* nfv = (const v2f*)nfeat;

  // strided group loop: wave id within relation = s*4+wave, stride BPR*4
  for (int g = s * 4 + wave; g < ngroups; g += BPR * 4) {
    const int base = bstart + g * 16;
    int count = bcnt - g * 16;
    if (count > 16) count = 16;

    // this lane's A-row (edge index within group) is c for BOTH halves
    const int e_a = (c < count) ? bin_edges[base + c] : -1;
    const int srow = (e_a >= 0) ? src[e_a] : 0;
    const int drow = (e_a >= 0) ? dst[e_a] : 0;

    v8f accT[4] = {};
    v8f accH[4] = {};
    // K loop: 16 steps of K=4. A frag: lane holds A[row=c, k=kb..kb+1].
    for (int t = 0; t < 16; ++t) {
      const int kb = 4 * t + 2 * h;
      v2f As = nfv[srow * (DFEAT / 2) + (kb >> 1)];
      v2f Ad = nfv[drow * (DFEAT / 2) + (kb >> 1)];
#pragma unroll
      for (int j = 0; j < 4; ++j) {
        v2f B;
        B.x = Ws[kb * DFEAT + 16 * j + c];
        B.y = Ws[(kb + 1) * DFEAT + 16 * j + c];
        accT[j] = wmma4(As, B, accT[j]);
        accH[j] = wmma4(Ad, B, accH[j]);
      }
    }

    // Epilogue: C/D element (vgpr i, lane) = [row i+8h, col 16j+c].
#pragma unroll
    for (int v = 0; v < 8; ++v) {
      const int e_m = __shfl(e_a, v + 8 * h, 32);  // lane v+8h holds row v+8h
      const int e_safe = (e_m < 0) ? 0 : e_m;
      float p = 0.0f;
#pragma unroll
      for (int j = 0; j < 4; ++j) {
        float tv = accT[j][v];
        float hv = accH[j][v];
        float ev = efeat[(size_t)e_safe * DFEAT + 16 * j + c];
        p += tv * tanhf(hv + ev);
      }
      p += __shfl_xor(p, 1, 32);
      p += __shfl_xor(p, 2, 32);
      p += __shfl_xor(p, 4, 32);
      p += __shfl_xor(p, 8, 32);
      if (c == 0 && e_m >= 0) {
        att[e_m] = p;
        atomicMax(&segkey[dst[e_m]], fkey(p));
      }
    }
  }
}

// ---------------------------------------------------------------- softmax pass
__global__ void softmax_kernel(const float* __restrict__ att,
                               const int* __restrict__ dst,
                               const unsigned* __restrict__ segkey,
                               float* __restrict__ ex,
                               float* __restrict__ denom, int E) {
  int e = blockIdx.x * blockDim.x + threadIdx.x;
  if (e >= E) return;
  int d = dst[e];
  float m = funkey(segkey[d]);
  float v = expf(att[e] - m);
  ex[e] = v;
  atomicAdd(&denom[d], v);
}

// ---------------------------------------------------------------- aggregation
// h_neighbor[dst] += a[e] * nfeat[src]; one wave per edge (64 floats).
__global__ __launch_bounds__(256) void aggregate_kernel(
    const float* __restrict__ nfeat, const int* __restrict__ src,
    const int* __restrict__ dst, const float* __restrict__ ex,
    const float* __restrict__ denom, float* __restrict__ hn, int E) {
  int wave = threadIdx.x >> 5;
  int lane = threadIdx.x & 31;
  int e = blockIdx.x * 8 + wave;
  if (e >= E) return;
  int d = dst[e];
  float a = ex[e] / denom[d];
  const v2f* nfv = (const v2f*)nfeat;
  v2f f = nfv[src[e] * (DFEAT / 2) + lane];
  atomicAdd(&hn[(size_t)d * DFEAT + 2 * lane], f.x * a);
  atomicAdd(&hn[(size_t)d * DFEAT + 2 * lane + 1], f.y * a);
}

// ---------------------------------------------------------------- output GEMMs
// out = lrelu((nfeat+hn)@W1) + lrelu((nfeat*hn)@W2); W1,W2 async-staged to LDS.
__global__ __launch_bounds__(256) void output_kernel(
    const float* __restrict__ nfeat, const float* __restrict__ hn,
    const float* __restrict__ W1, const float* __restrict__ W2,
    float* __restrict__ out, int N) {
  __shared__ float W1s[DFEAT * DFEAT];
  __shared__ float W2s[DFEAT * DFEAT];
  {
    const unsigned b1 = lds_offset(&W1s[0]);
    const unsigned b2 = lds_offset(&W2s[0]);
    for (int i = threadIdx.x; i < (DFEAT * DFEAT) / 4; i += 256) {
      async_load_b128(W1 + i * 4, b1 + i * 16);
      async_load_b128(W2 + i * 4, b2 + i * 16);
    }
    wait_asynccnt0();
  }
  __syncthreads();

  const int wave = threadIdx.x >> 5;
  const int lane = threadIdx.x & 31;
  const int h = lane >> 4;
  const int c = lane & 15;
  const int m0 = blockIdx.x * 128 + wave * 16;  // 16-row tile per wave
  const int myrow = m0 + c;
  const int rowc = (myrow < N) ? myrow : 0;

  const v2f* nfv = (const v2f*)nfeat;
  const v2f* hnv = (const v2f*)hn;

  v8f acc1[4] = {};
  v8f acc2[4] = {};
  for (int t = 0; t < 16; ++t) {
    const int kb = 4 * t + 2 * h;
    v2f nf = nfv[rowc * (DFEAT / 2) + (kb >> 1)];
    v2f hv = hnv[rowc * (DFEAT / 2) + (kb >> 1)];
    v2f As = nf + hv;   // (nfeat + h_neighbor)
    v2f Ap = nf * hv;   // (nfeat * h_neighbor)
#pragma unroll
    for (int j = 0; j < 4; ++j) {
      v2f B1, B2;
      B1.x = W1s[kb * DFEAT + 16 * j + c];
      B1.y = W1s[(kb + 1) * DFEAT + 16 * j + c];
      B2.x = W2s[kb * DFEAT + 16 * j + c];
      B2.y = W2s[(kb + 1) * DFEAT + 16 * j + c];
      acc1[j] = wmma4(As, B1, acc1[j]);
      acc2[j] = wmma4(Ap, B2, acc2[j]);
    }
  }

#pragma unroll
  for (int j = 0; j < 4; ++j) {
#pragma unroll
    for (int i = 0; i < 8; ++i) {
      int m = m0 + i + 8 * h;
      if (m < N) {
        float x1 = acc1[j][i];
        float x2 = acc2[j][i];
        float v = (x1 >= 0.0f ? x1 : 0.01f * x1) +
                  (x2 >= 0.0f ? x2 : 0.01f * x2);
        out[(size_t)m * DFEAT + 16 * j + c] = v;
      }
    }
  }
}

// ---------------------------------------------------------------- launcher
extern "C" void kernel_launch(void* const* d_in, const int* in_sizes, int n_in,
                              void* d_out, int out_size, void* d_ws,
                              size_t ws_size, hipStream_t stream) {
  const float* nfeat = (const float*)d_in[0];
  const float* efeat = (const float*)d_in[1];
  const float* relW = (const float*)d_in[2];
  const float* W1 = (const float*)d_in[3];
  const float* W2 = (const float*)d_in[4];
  const int* src = (const int*)d_in[5];
  const int* dst = (const int*)d_in[6];
  const int* ety = (const int*)d_in[7];

  const int N = in_sizes[0] / DFEAT;                 // 50000
  const int E = in_sizes[5];                         // 800000
  const int R = in_sizes[2] / (DFEAT * DFEAT);       // 16

  // workspace layout (4-byte words)
  int* cnt = (int*)d_ws;                    // [R]
  int* bin_start = cnt + MAXREL;            // [R+1]
  int* cursor = bin_start + MAXREL + 1;     // [R]
  unsigned* segkey = (unsigned*)(cnt + 128);          // [N]
  float* denom = (float*)(segkey + N);                // [N]
  float* att = denom + N;                             // [E]
  float* ex = att + E;                                // [E]
  int* bin_edges = (int*)(ex + E);                    // [E]

  float* hn = (float*)d_out;                       // output 0: h_neighbor
  float* out = hn + (size_t)N * DFEAT;             // output 1: out

  const long initTotal = (long)N * DFEAT;
  init_kernel<<<(int)((initTotal + 255) / 256), 256, 0, stream>>>(
      hn, denom, segkey, cnt, N, R, initTotal);

  hist_kernel<<<(E + 255) / 256, 256, 0, stream>>>(ety, cnt, E);
  prefix_kernel<<<1, 32, 0, stream>>>(cnt, bin_start, cursor, R);
  scatter_kernel<<<(E + 255) / 256, 256, 0, stream>>>(ety, cursor, bin_edges, E);

  edge_att_kernel<<<R * BPR, 128, 0, stream>>>(nfeat, efeat, relW, src, dst,
                                               bin_edges, bin_start, cnt, att,
                                               segkey);

  softmax_kernel<<<(E + 255) / 256, 256, 0, stream>>>(att, dst, segkey, ex,
                                                      denom, E);

  aggregate_kernel<<<(E + 7) / 8, 256, 0, stream>>>(nfeat, src, dst, ex, denom,
                                                    hn, E);

  output_kernel<<<(N + 127) / 128, 256, 0, stream>>>(nfeat, hn, W1, W2, out, N);
}